// QuantumLayer_88596585382720
// MI455X (gfx1250) — compile-verified
//
#include <hip/hip_runtime.h>
#include <hip/hip_bf16.h>
#include <math.h>
#include <stdint.h>

// Quantum circuit simulator: 20 qubits, batch 16, 4 layers of (RY x20, CNOT chain).
//  * Circuit is entirely real -> fp32 simulation; final abs == fabsf.
//  * 4 RY gates fuse into one 16x16 Kronecker matrix -> V_WMMA_F32_16X16X4_F32 (x4 K-chunks).
//  * CNOT chain q=0..18 == Gray permutation new[i] = old[i ^ (i>>1)], fused into the next
//    layer's gather; the LAST chain + |.| is fused into layer-3 pass B as an inverse-Gray
//    scatter (stays within aligned 16-float blocks -> full 64B lines).
//  * 8 full-state streaming passes: ~1.0 GB @ 23.3 TB/s ~= 44 us. HBM-bound.
//  * Pass A tile I/O uses GLOBAL_LOAD_ASYNC_TO_LDS_B128 / STORE_ASYNC_FROM_LDS_B128
//    (ASYNCcnt DMA, no VGPR staging) when the builtins exist, else float4 copies.
// Requires ws_size >= 64MB + 32KB (state scratch + 20 fused 16x16 matrices).

typedef __attribute__((ext_vector_type(2))) float v2f;
typedef __attribute__((ext_vector_type(4))) float v4f;
typedef __attribute__((ext_vector_type(8))) float v8f;
typedef __attribute__((ext_vector_type(4))) int v4i;

#define NQ 20
#define NL 4
#define NBATCH 16

#if defined(__gfx1250__) && __has_builtin(__builtin_amdgcn_global_load_async_to_lds_b128) && \
    __has_builtin(__builtin_amdgcn_global_store_async_from_lds_b128)
#define USE_ASYNC_LDS 1
typedef __attribute__((address_space(1))) v4i as1_v4i;  // global int4
typedef __attribute__((address_space(3))) v4i as3_v4i;  // LDS int4
#else
#define USE_ASYNC_LDS 0
#endif

__device__ __forceinline__ void wait_async0() {
#if __has_builtin(__builtin_amdgcn_s_wait_asynccnt)
  __builtin_amdgcn_s_wait_asynccnt(0);
#else
  asm volatile("s_wait_asynccnt 0x0" ::: "memory");
#endif
}

// ---------------------------------------------------------------------------
// Build the 20 fused 16x16 group matrices: U[l][g] = RY(t_{4g}) (x) ... (x) RY(t_{4g+3}).
// Group-index bit3 corresponds to qubit 4g (most significant of the group).
__global__ __launch_bounds__(256) void compute_u_kernel(const float* __restrict__ theta,
                                                        float* __restrict__ U) {
  int T = threadIdx.x;
  int i = T >> 4;
  int j = T & 15;
  for (int lg = 0; lg < NL * 5; ++lg) {
    int l = lg / 5, g = lg % 5;
    float val = 1.0f;
#pragma unroll
    for (int k = 0; k < 4; ++k) {
      float th = theta[l * NQ + g * 4 + k] * 0.5f;
      float c = cosf(th), s = sinf(th);
      int r = (i >> (3 - k)) & 1;
      int cc = (j >> (3 - k)) & 1;
      // RY = [[c, -s], [s, c]]
      val *= (r == cc) ? c : ((r == 0) ? -s : s);
    }
    U[lg * 256 + T] = val;
  }
}

// ---------------------------------------------------------------------------
// Apply one 16x16 group matrix to a 4096-float LDS tile along the nibble at
// `shift`. One wave handles 32 of the 256 columns (2 WMMA column-tiles).
// WMMA fp32 layouts (wave32):
//   A 16x4 : lanes 0-15 vgpr{0,1} = K{0,1}; lanes 16-31 = K{2,3}; M = lane&15
//   B 4x16 : mirrored; N = lane&15
//   C/D    : vgpr v -> M = v + 8*(lane>>4), N = lane&15
__device__ __forceinline__ void apply_group(float* lds, const float* __restrict__ Ug,
                                            int shift, int wave, int l16, int h) {
  v2f afrag[4];
#pragma unroll
  for (int k = 0; k < 4; ++k) {
    afrag[k].x = Ug[l16 * 16 + 4 * k + 2 * h];
    afrag[k].y = Ug[l16 * 16 + 4 * k + 2 * h + 1];
  }
  int mask = (1 << shift) - 1;
#pragma unroll
  for (int ct = 0; ct < 2; ++ct) {
    int col = wave * 32 + ct * 16 + l16;
    int colPart = ((col >> shift) << (shift + 4)) | (col & mask);
    v8f acc = {0.f, 0.f, 0.f, 0.f, 0.f, 0.f, 0.f, 0.f};
#pragma unroll
    for (int k = 0; k < 4; ++k) {
      int r0 = 4 * k + 2 * h;
      v2f bfrag;
      bfrag.x = lds[colPart | (r0 << shift)];
      bfrag.y = lds[colPart | ((r0 + 1) << shift)];
      acc = __builtin_amdgcn_wmma_f32_16x16x4_f32(false, afrag[k], false, bfrag,
                                                  (short)0, acc, false, false);
    }
#pragma unroll
    for (int v = 0; v < 8; ++v) {
      lds[colPart | ((v + 8 * h) << shift)] = acc[v];
    }
  }
}

// ---------------------------------------------------------------------------
// Pass A: contiguous 4096-float tile (bit positions 0..11). Applies groups
// 2,3,4 (qubits 8..19). Optionally fuses the previous layer's CNOT chain
// (Gray gather): src = idx ^ (idx>>1) stays inside one aligned 4096 block
// for the low 12 bits (and inside aligned 16-blocks for the low 4 -> full lines).
__global__ __launch_bounds__(256) void pass_a_kernel(const float* __restrict__ in,
                                                     float* __restrict__ out,
                                                     const float* __restrict__ Ubase,
                                                     int doGray) {
  __shared__ float lds[4096];
  int T = threadIdx.x;
  int blk = blockIdx.x;
  int b = blk >> 8;
  int tile = blk & 255;  // bit positions 12..19
  size_t outBase = ((size_t)b << NQ) | ((size_t)tile << 12);

  if (doGray) {
    int srcTile = tile ^ (tile >> 1);
    int xor11 = (tile & 1) << 11;  // src bit11 = t11 ^ tile bit0
    size_t inBase = ((size_t)b << NQ) | ((size_t)srcTile << 12);
#pragma unroll
    for (int i = 0; i < 16; ++i) {
      int t = T + 256 * i;
      int srcLow = (t ^ (t >> 1)) ^ xor11;
      lds[t] = in[inBase + (size_t)srcLow];
    }
  } else {
#if USE_ASYNC_LDS
#pragma unroll
    for (int i = 0; i < 4; ++i) {
      int f4 = T + 256 * i;  // float4 index within tile
      __builtin_amdgcn_global_load_async_to_lds_b128(
          (as1_v4i*)(uintptr_t)(in + outBase + (size_t)f4 * 4),
          (as3_v4i*)(uintptr_t)(&lds[f4 * 4]), 0, 0);
    }
    wait_async0();
#else
    const v4f* __restrict__ in4 = (const v4f*)(in + outBase);
    v4f* lds4 = (v4f*)lds;
#pragma unroll
    for (int i = 0; i < 4; ++i) {
      int f4 = T + 256 * i;
      lds4[f4] = in4[f4];
    }
#endif
  }
  __syncthreads();

  int wave = T >> 5, lane = T & 31, l16 = lane & 15, h = lane >> 4;
  for (int g = 2; g < 5; ++g) {  // shifts 8, 4, 0
    apply_group(lds, Ubase + g * 256, (4 - g) * 4, wave, l16, h);
    __syncthreads();
  }

#if USE_ASYNC_LDS
#pragma unroll
  for (int i = 0; i < 4; ++i) {
    int f4 = T + 256 * i;
    __builtin_amdgcn_global_store_async_from_lds_b128(
        (as1_v4i*)(uintptr_t)(out + outBase + (size_t)f4 * 4),
        (as3_v4i*)(uintptr_t)(&lds[f4 * 4]), 0, 0);
  }
  wait_async0();
#else
  v4f* __restrict__ out4 = (v4f*)(out + outBase);
  v4f* lds4w = (v4f*)lds;
#pragma unroll
  for (int i = 0; i < 4; ++i) {
    int f4 = T + 256 * i;
    out4[f4] = lds4w[f4];
  }
#endif
}

// ---------------------------------------------------------------------------
// Pass B: strided tile covering bit positions 12..19 (groups 0,1 = qubits
// 0..7). Each row provides 16 contiguous floats -> b128 per lane, 4 lanes per
// 64B line. finalMode==1: fuse layer-3 CNOT chain + |.| as an inverse-Gray
// scatter into `fin` (bits>=4 of invgray are fixed per row -> full lines).
__global__ __launch_bounds__(256) void pass_b_kernel(float* __restrict__ data,
                                                     float* __restrict__ fin,
                                                     const float* __restrict__ Ubase,
                                                     int finalMode) {
  __shared__ float lds[4096];
  int T = threadIdx.x;
  int blk = blockIdx.x;
  int b = blk >> 8;
  int tile = blk & 255;  // low-address block (bit positions 4..11)
  size_t base = ((size_t)b << NQ) | ((size_t)tile << 4);
  int lo4 = (T & 3) << 2;  // float4 offset within the 16-float row chunk
  int u0 = T >> 2;         // 0..63, bit positions 12..19
  v4f* lds4 = (v4f*)lds;

#pragma unroll
  for (int i = 0; i < 4; ++i) {
    int u = u0 + 64 * i;
    lds4[((u << 4) | lo4) >> 2] = *(const v4f*)(data + base + ((size_t)u << 12) + lo4);
  }
  __syncthreads();

  int wave = T >> 5, lane = T & 31, l16 = lane & 15, h = lane >> 4;
  // LDS index e = (u<<4)|lo : group0 -> e bits 8..11 (shift 8), group1 -> bits 4..7 (shift 4)
  for (int g = 0; g < 2; ++g) {
    apply_group(lds, Ubase + g * 256, 8 - 4 * g, wave, l16, h);
    __syncthreads();
  }

  if (finalMode) {
#pragma unroll
    for (int i = 0; i < 4; ++i) {
      int u = u0 + 64 * i;
      v4f v = lds4[((u << 4) | lo4) >> 2];
#pragma unroll
      for (int c = 0; c < 4; ++c) {
        int e = (u << 12) | (tile << 4) | (lo4 + c);  // 20-bit state index
        int ig = e ^ (e >> 1);                        // inverse Gray (prefix XOR)
        ig ^= ig >> 2;
        ig ^= ig >> 4;
        ig ^= ig >> 8;
        ig ^= ig >> 16;
        float val = (c == 0) ? v.x : (c == 1) ? v.y : (c == 2) ? v.z : v.w;
        fin[((size_t)b << NQ) | (size_t)ig] = fabsf(val);
      }
    }
  } else {
#pragma unroll
    for (int i = 0; i < 4; ++i) {
      int u = u0 + 64 * i;
      *(v4f*)(data + base + ((size_t)u << 12) + lo4) = lds4[((u << 4) | lo4) >> 2];
    }
  }
}

// ---------------------------------------------------------------------------
extern "C" void kernel_launch(void* const* d_in, const int* in_sizes, int n_in,
                              void* d_out, int out_size, void* d_ws, size_t ws_size,
                              hipStream_t stream) {
  (void)in_sizes; (void)n_in; (void)out_size; (void)ws_size;
  const float* x = (const float*)d_in[0];      // (16, 2^20) fp32
  const float* theta = (const float*)d_in[1];  // (80,) fp32
  float* out = (float*)d_out;                  // (16, 2^20) fp32
  float* U = (float*)d_ws;                     // 20 * 256 floats (20 KB)
  float* W = (float*)((char*)d_ws + 32768);    // 64 MB state scratch

  const int nblk = NBATCH * 256;  // 4096 blocks per streaming pass

  compute_u_kernel<<<1, 256, 0, stream>>>(theta, U);

  // Layer 0 (no preceding CNOT chain)
  pass_a_kernel<<<nblk, 256, 0, stream>>>(x, out, U + 0 * 1280, 0);
  pass_b_kernel<<<nblk, 256, 0, stream>>>(out, nullptr, U + 0 * 1280, 0);
  // Layer 1 (layer-0 CNOT chain fused into the gather)
  pass_a_kernel<<<nblk, 256, 0, stream>>>(out, W, U + 1 * 1280, 1);
  pass_b_kernel<<<nblk, 256, 0, stream>>>(W, nullptr, U + 1 * 1280, 0);
  // Layer 2
  pass_a_kernel<<<nblk, 256, 0, stream>>>(W, out, U + 2 * 1280, 1);
  pass_b_kernel<<<nblk, 256, 0, stream>>>(out, nullptr, U + 2 * 1280, 0);
  // Layer 3 (pass B also applies the final CNOT chain + |.| as inverse-Gray scatter)
  pass_a_kernel<<<nblk, 256, 0, stream>>>(out, W, U + 3 * 1280, 1);
  pass_b_kernel<<<nblk, 256, 0, stream>>>(W, out, U + 3 * 1280, 1);
}